// SmoothPrec_at_K_Loss_55817394979147
// MI455X (gfx1250) — compile-verified
//
#include <hip/hip_runtime.h>
#include <hip/hip_bf16.h>
#include <float.h>
#include <math.h>

// ---------------------------------------------------------------------------
// SmoothPrec@K loss, MI455X (gfx1250, wave32).
//
// Dead-code analysis of the reference:
//   * err_pos is the constant 2K/N (top_k indices are always > -1).
//   * sparsity is the constant 0.
//   * loss only uses score_hat[-1]  => only batch b = 63 matters.
// Real work: A63(33x128) x X63^T(128x8192) fp32 GEMM (V_WMMA_F32_16X16X4_F32),
// fused epilogue (normalize, diagonal-skip remap, margin), then per-row
// top-17 + log-domain elementary-symmetric-polynomial DP (parallelized via
// associative e_k merge = log-conv of 17-vectors).
// ---------------------------------------------------------------------------

typedef __attribute__((ext_vector_type(2))) float v2f;
typedef __attribute__((ext_vector_type(8))) float v8f;

#define B_     64
#define NCOL   8192          // N+1
#define NSC    8191          // N
#define DDIM   128
#define PP     33
#define KK     16
#define NEGF   (-1.0e30f)
#define TOPNEG (-3.4e38f)
#define HARD_T 73.682724f    // K * TAU * float32(log(100))

// workspace layout (float indices)
#define WS_GSS   0                      // 8192
#define WS_ANORM 8192                   // 33 (pad 48)
#define WS_CAT   8240                   // 1 int
#define WS_X1    8448                   // 33 rows, stride NCOL
#define WS_V1    (WS_X1 + PP * NCOL)    // 33
#define WS_V2    (WS_V1 + 64)           // 33

__device__ __forceinline__ float lae(float a, float b) {
  float m = fmaxf(a, b);
  float d = fabsf(a - b);
  return m + log1pf(__expf(-d));
}

// ---- kernel A: gss[n] = ||X63[n]||^2, anorm[p] = ||A63[p]||, categories ----
__global__ void k_pre(const float* __restrict__ x, const int* __restrict__ y,
                      float* __restrict__ ws) {
  const int tid = threadIdx.x;
  const size_t base = (size_t)63 * NCOL * DDIM;
  if (blockIdx.x < 32) {
    int n = blockIdx.x * 256 + tid;
    const float4* row = (const float4*)(x + base + (size_t)n * DDIM);
    float s = 0.f;
#pragma unroll
    for (int i = 0; i < DDIM / 4; ++i) {
      float4 v = row[i];
      s += v.x * v.x + v.y * v.y + v.z * v.z + v.w * v.w;
    }
    ws[WS_GSS + n] = s;
  } else {
    if (tid < PP) {
      const float4* row = (const float4*)(x + base + (size_t)tid * DDIM);
      float s = 0.f;
#pragma unroll
      for (int i = 0; i < DDIM / 4; ++i) {
        float4 v = row[i];
        s += v.x * v.x + v.y * v.y + v.z * v.z + v.w * v.w;
      }
      ws[WS_ANORM + tid] = sqrtf(s);
    }
    __shared__ int cnt;
    if (tid == 0) cnt = 0;
    __syncthreads();
    const int* y63 = y + 63 * NCOL;
    int head = y63[0];
    int local = 0;
    for (int i = tid; i < NCOL; i += 256) local += (y63[i] == head) ? 1 : 0;
    atomicAdd(&cnt, local);
    __syncthreads();
    if (tid == 0) ((int*)ws)[WS_CAT] = cnt - 1;
  }
}

// ---- kernel B: WMMA fp32 GEMM + fused score_hat epilogue --------------------
// grid 64 x 256 (8 waves). Each wave: one 16-col N-tile, 3 M-tiles (48 rows).
__global__ void __launch_bounds__(256) k_gemm(const float* __restrict__ x,
                                              float* __restrict__ ws) {
  __shared__ float As[48 * DDIM];   // rows 0..32 valid; 33..47 junk (discarded)
  __shared__ float anorm_s[PP];
  __shared__ int cat_s;
  const int tid = threadIdx.x;
  const size_t base = (size_t)63 * NCOL * DDIM;

  for (int i = tid; i < PP * DDIM; i += 256) As[i] = x[base + i];
  if (tid < PP) anorm_s[tid] = ws[WS_ANORM + tid];
  if (tid == 0) cat_s = ((const int*)ws)[WS_CAT];
  __syncthreads();

  const int wid = tid >> 5, lane = tid & 31;
  const int nn = lane & 15, h = lane >> 4;
  const int n = blockIdx.x * 128 + wid * 16 + nn;
  const float2* Xr = (const float2*)(x + base + (size_t)n * DDIM);
  const float2* As2 = (const float2*)As;
  const float gssn = ws[WS_GSS + n];
  const int cat = cat_s;

  v8f acc0 = {0.f, 0.f, 0.f, 0.f, 0.f, 0.f, 0.f, 0.f};
  v8f acc1 = acc0, acc2 = acc0;

#pragma unroll 4
  for (int s = 0; s < 32; ++s) {
    // B fragment: B[k][nc] = X[n0+nc][4s+k]; lane holds k = 2h, 2h+1
    float2 bv2 = Xr[2 * s + h];
    v2f b; b.x = bv2.x; b.y = bv2.y;
    // A fragments: A[m][k]; lane holds row m = t*16+nn, k = 2h, 2h+1
    float2 a0v = As2[(0 * 16 + nn) * (DDIM / 2) + 2 * s + h];
    float2 a1v = As2[(1 * 16 + nn) * (DDIM / 2) + 2 * s + h];
    float2 a2v = As2[(2 * 16 + nn) * (DDIM / 2) + 2 * s + h];
    v2f a0; a0.x = a0v.x; a0.y = a0v.y;
    v2f a1; a1.x = a1v.x; a1.y = a1v.y;
    v2f a2; a2.x = a2v.x; a2.y = a2v.y;
    acc0 = __builtin_amdgcn_wmma_f32_16x16x4_f32(false, a0, false, b, (short)0,
                                                 acc0, false, false);
    acc1 = __builtin_amdgcn_wmma_f32_16x16x4_f32(false, a1, false, b, (short)0,
                                                 acc1, false, false);
    acc2 = __builtin_amdgcn_wmma_f32_16x16x4_f32(false, a2, false, b, (short)0,
                                                 acc2, false, false);
  }

  // Epilogue: C/D layout: lanes 0-15: M = r, lanes 16-31: M = 8 + r; N = lane&15.
  const int mbase = (lane >= 16) ? 8 : 0;
  float* x1 = ws + WS_X1;
  auto emit = [&](float val, int p) {
    if (p < PP) {
      float sf = val / (anorm_s[p] * gssn);
      if (n != p) {
        int c = n - ((n > p) ? 1 : 0);
        x1[p * NCOL + c] = sf + ((c >= cat) ? 1.0f : 0.0f);
      }
    }
  };
#pragma unroll
  for (int r = 0; r < 8; ++r) {
    emit(acc0[r], 0 + mbase + r);
    emit(acc1[r], 16 + mbase + r);
    emit(acc2[r], 32 + mbase + r);
  }
}

// ---- kernel C: per-row top-17 + log e_k over 8191 values --------------------
__global__ void __launch_bounds__(256) k_row(float* __restrict__ ws) {
  __shared__ float Ebuf[256 * 17];
  __shared__ float rv[256];
  __shared__ int rt[256];
  const int tid = threadIdx.x;
  const int p = blockIdx.x;
  const float* row = ws + WS_X1 + p * NCOL;

  float loc[17];
#pragma unroll
  for (int j = 0; j < 17; ++j) loc[j] = TOPNEG;
  float L[17];
  L[0] = 0.f;
#pragma unroll
  for (int j = 1; j < 17; ++j) L[j] = NEGF;

  for (int c = tid; c < NSC; c += 256) {
    float v = row[c];
    if (v > loc[16]) {                 // local sorted top-17 (descending)
      loc[16] = v;
#pragma unroll
      for (int j = 16; j > 0; --j) {
        if (loc[j] > loc[j - 1]) { float t = loc[j - 1]; loc[j - 1] = loc[j]; loc[j] = t; }
      }
    }
#pragma unroll
    for (int j = 16; j >= 1; --j) L[j] = lae(L[j], L[j - 1] + v);
  }

  // ---- block top-17 merge: 17 rounds of argmax-pop ----
#pragma unroll
  for (int j = 0; j < 17; ++j) Ebuf[tid * 17 + j] = loc[j];
  __syncthreads();
  int ptr = 0;
  float hsum = 0.f, t15 = 0.f, t16 = 0.f;
  for (int round = 0; round < 17; ++round) {
    rv[tid] = (ptr < 17) ? Ebuf[tid * 17 + ptr] : TOPNEG;
    rt[tid] = tid;
    __syncthreads();
    for (int off = 128; off > 0; off >>= 1) {
      if (tid < off) {
        float o = rv[tid + off]; int ot = rt[tid + off];
        if (o > rv[tid] || (o == rv[tid] && ot < rt[tid])) { rv[tid] = o; rt[tid] = ot; }
      }
      __syncthreads();
    }
    float w = rv[0]; int wt = rt[0];
    __syncthreads();
    if (tid == wt) ptr++;
    if (round < 16) hsum += w;
    if (round == 15) t15 = w;
    if (round == 16) t16 = w;
  }
  __syncthreads();

  // ---- e_k tree combine: log-domain 17-vector convolution ----
#pragma unroll
  for (int j = 0; j < 17; ++j) Ebuf[tid * 17 + j] = L[j];
  __syncthreads();
  for (int off = 128; off > 0; off >>= 1) {
    if (tid < off) {
      float a[17], b[17], o[17];
#pragma unroll
      for (int j = 0; j < 17; ++j) {
        a[j] = Ebuf[tid * 17 + j];
        b[j] = Ebuf[(tid + off) * 17 + j];
      }
#pragma unroll
      for (int k = 0; k < 17; ++k) {
        float r = a[0] + b[k];
#pragma unroll
        for (int i = 1; i <= k; ++i) r = lae(r, a[i] + b[k - i]);
        o[k] = r;
      }
#pragma unroll
      for (int j = 0; j < 17; ++j) Ebuf[tid * 17 + j] = o[j];
    }
    __syncthreads();
  }
  if (tid == 0) {
    float fsk = Ebuf[16];                       // TAU * L[K], TAU = 1
    bool hard = (t15 - t16) >= HARD_T;
    ws[WS_V1 + p] = hard ? hsum : fsk;
  }
}

// ---- kernel D: x2 = first 32 columns of each row (serial per row, tiny) -----
__global__ void k_x2(float* __restrict__ ws) {
  int p = threadIdx.x;
  if (p >= PP) return;
  const float* row = ws + WS_X1 + p * NCOL;
  float loc[17];
#pragma unroll
  for (int j = 0; j < 17; ++j) loc[j] = TOPNEG;
  float L[17];
  L[0] = 0.f;
#pragma unroll
  for (int j = 1; j < 17; ++j) L[j] = NEGF;
  for (int c = 0; c < PP - 1; ++c) {
    float v = row[c];
    if (v > loc[16]) {
      loc[16] = v;
#pragma unroll
      for (int j = 16; j > 0; --j) {
        if (loc[j] > loc[j - 1]) { float t = loc[j - 1]; loc[j - 1] = loc[j]; loc[j] = t; }
      }
    }
#pragma unroll
    for (int j = 16; j >= 1; --j) L[j] = lae(L[j], L[j - 1] + v);
  }
  float hsum = 0.f;
#pragma unroll
  for (int j = 0; j < 16; ++j) hsum += loc[j];
  bool hard = (loc[15] - loc[16]) >= HARD_T;
  ws[WS_V2 + p] = hard ? hsum : L[16];
}

// ---- kernel E: finalize (loss, sparsity, err_pos) ---------------------------
__global__ void k_fin(const float* __restrict__ ws, float* __restrict__ out) {
  __shared__ float s1[64], s2[64];
  int t = threadIdx.x;
  s1[t] = (t < PP) ? ws[WS_V1 + t] : 0.f;
  s2[t] = (t < PP) ? ws[WS_V2 + t] : 0.f;
  __syncthreads();
  for (int off = 32; off > 0; off >>= 1) {
    if (t < off) { s1[t] += s1[t + off]; s2[t] += s2[t + off]; }
    __syncthreads();
  }
  if (t == 0) {
    out[0] = (s1[0] - s2[0]) / (float)PP;
    out[1] = 0.f;
    out[2] = (float)(2.0 * (double)(B_ * PP * KK) / (double)((double)B_ * PP * NSC));
  }
}

extern "C" void kernel_launch(void* const* d_in, const int* in_sizes, int n_in,
                              void* d_out, int out_size, void* d_ws, size_t ws_size,
                              hipStream_t stream) {
  const float* x = (const float*)d_in[0];
  const int* y = (const int*)d_in[1];
  float* ws = (float*)d_ws;
  float* out = (float*)d_out;
  (void)in_sizes; (void)n_in; (void)out_size; (void)ws_size;

  k_pre<<<33, 256, 0, stream>>>(x, y, ws);
  k_gemm<<<64, 256, 0, stream>>>(x, ws);
  k_row<<<PP, 256, 0, stream>>>(ws);
  k_x2<<<1, 64, 0, stream>>>(ws);
  k_fin<<<1, 64, 0, stream>>>(ws, out);
}